// Circular_Spline_Flow_20572893348083
// MI455X (gfx1250) — compile-verified
//
#include <hip/hip_runtime.h>
#include <math.h>

typedef float v2f __attribute__((ext_vector_type(2)));
typedef float v8f __attribute__((ext_vector_type(8)));

#define TWO_PI_F     6.2831853071795864769f
#define MIN_W        1e-3f
#define MIN_H        1e-3f
#define MIN_D        1e-3f
#define DERIV_SHIFT  0.54132485461292f   /* log(e-1): softplus(x+shift)=1 at x=0 */

#define KBINS 32
#define DH    8
#define CDIM  64
#define HDIM  256
#define NPAR  768            /* 3*K*Dh */

#define WAVES           2
#define MTILES          2                      /* 16-row M tiles per wave   */
#define ROWS_PER_WAVE   (MTILES * 16)          /* 32                        */
#define ROWS_PER_BLOCK  (WAVES * ROWS_PER_WAVE)/* 64                        */
#define PCHUNK          96   /* param columns per chunk = one head (3K=96)  */
#define NT2             (PCHUNK / 16)          /* 6 N-tiles per chunk       */

/* dynamic LDS: hbuf 64*256 f32 (64KB) + pbuf 64*96 f32 (24KB) = 88KB */
#define HBUF_FLOATS (ROWS_PER_BLOCK * HDIM)
#define PBUF_FLOATS (ROWS_PER_BLOCK * PCHUNK)
#define LDS_BYTES   ((HBUF_FLOATS + PBUF_FLOATS) * 4)

static __device__ __forceinline__ float softplus_f(float x) {
    return (x > 20.0f) ? x : log1pf(__expf(x));
}

__global__ __launch_bounds__(WAVES * 32)
void Circular_Spline_Flow_fused_kernel(
    const float* __restrict__ theta,   // (B, 8)
    const float* __restrict__ x,       // (B, 64)
    const float* __restrict__ W1,      // (64, 256) row-major
    const float* __restrict__ b1,      // (256)
    const float* __restrict__ W2,      // (256, 768) row-major
    const float* __restrict__ b2,      // (768)
    const float* __restrict__ eta,     // (1)
    float* __restrict__ out,           // outputs (B*8) then logabsdet (B*8)
    int Btotal)
{
    extern __shared__ float smem[];

    const int tid  = threadIdx.x;
    const int wave = tid >> 5;
    const int lane = tid & 31;
    const int m    = lane & 15;   // row-in-tile / B-frag column
    const int half = lane >> 4;   // K sub-group for 16x16x4 fragments

    const int rowbase = blockIdx.x * ROWS_PER_BLOCK + wave * ROWS_PER_WAVE;

    float* hbuf = smem + wave * ROWS_PER_WAVE * HDIM;                // 32x256
    float* pbuf = smem + HBUF_FLOATS + wave * ROWS_PER_WAVE * PCHUNK;// 32x96

    const float etav = eta[0];

    // ======================= GEMM1: H = relu(X @ W1 + b1) =======================
    // A-frag layout for V_WMMA_F32_16X16X4_F32: lane(l<16): M=l, K={0,1};
    // lane(l>=16): M=l-16, K={2,3}. Hoist both 16x64 X slices into registers.
    v2f afrag[MTILES][16];
    #pragma unroll
    for (int mt = 0; mt < MTILES; ++mt) {
        const float* xrow = x + (size_t)(rowbase + mt * 16 + m) * CDIM;
        #pragma unroll
        for (int kc = 0; kc < 16; ++kc) {
            const int kk = 4 * kc + 2 * half;
            afrag[mt][kc].x = xrow[kk];
            afrag[mt][kc].y = xrow[kk + 1];
        }
    }

    for (int j = 0; j < 16; ++j) {          // 16 N-tiles -> 256 cols of H
        const int col = 16 * j + m;
        v8f acc0 = {0.f,0.f,0.f,0.f,0.f,0.f,0.f,0.f};
        v8f acc1 = {0.f,0.f,0.f,0.f,0.f,0.f,0.f,0.f};
        #pragma unroll
        for (int kc = 0; kc < 16; ++kc) {
            const int kk = 4 * kc + 2 * half;
            v2f bfrag;
            bfrag.x = W1[(size_t)kk * HDIM + col];
            bfrag.y = W1[(size_t)(kk + 1) * HDIM + col];
            acc0 = __builtin_amdgcn_wmma_f32_16x16x4_f32(
                false, afrag[0][kc], false, bfrag, (short)0, acc0, false, false);
            acc1 = __builtin_amdgcn_wmma_f32_16x16x4_f32(
                false, afrag[1][kc], false, bfrag, (short)0, acc1, false, false);
        }
        const float bias = b1[col];
        #pragma unroll
        for (int v = 0; v < 8; ++v) {       // D elem v -> row M = v + 8*half
            float h0 = fmaxf(acc0[v] + bias, 0.0f);
            float h1 = fmaxf(acc1[v] + bias, 0.0f);
            hbuf[( 0 + 8 * half + v) * HDIM + col] = h0;
            hbuf[(16 + 8 * half + v) * HDIM + col] = h1;
        }
    }
    // waves only touch their own LDS slices; same-wave DS ops are in-order,
    // so no barrier is required before consuming hbuf.

    // ============== GEMM2 (chunked) + fused circular RQ spline ==============
    // One chunk = 96 params = one head's (uw,uh,ud). K-outer loop hoists the
    // A-fragments (shared by all 6 N-tiles); 12 live accumulators (96 VGPRs).
    for (int dp = 0; dp < DH; ++dp) {
        const int colbase = dp * PCHUNK;

        // warm L2 for this chunk's W2 columns (global_prefetch_b8)
        __builtin_prefetch(&W2[(size_t)(2 * half) * NPAR + colbase + m], 0, 1);

        v8f acc[MTILES][NT2];
        #pragma unroll
        for (int mt = 0; mt < MTILES; ++mt)
            #pragma unroll
            for (int jt = 0; jt < NT2; ++jt)
                acc[mt][jt] = (v8f){0.f,0.f,0.f,0.f,0.f,0.f,0.f,0.f};

        #pragma unroll 2
        for (int kc = 0; kc < 64; ++kc) {   // K = 256 in steps of 4
            const int kk = 4 * kc + 2 * half;
            v2f a0, a1;                      // ds_load_b64 each
            a0.x = hbuf[( 0 + m) * HDIM + kk];
            a0.y = hbuf[( 0 + m) * HDIM + kk + 1];
            a1.x = hbuf[(16 + m) * HDIM + kk];
            a1.y = hbuf[(16 + m) * HDIM + kk + 1];
            #pragma unroll
            for (int jt = 0; jt < NT2; ++jt) {
                const int col = colbase + 16 * jt + m;
                v2f b;
                b.x = W2[(size_t)kk * NPAR + col];
                b.y = W2[(size_t)(kk + 1) * NPAR + col];
                acc[0][jt] = __builtin_amdgcn_wmma_f32_16x16x4_f32(
                    false, a0, false, b, (short)0, acc[0][jt], false, false);
                acc[1][jt] = __builtin_amdgcn_wmma_f32_16x16x4_f32(
                    false, a1, false, b, (short)0, acc[1][jt], false, false);
            }
        }

        // epilogue: bias + eta scale -> pbuf (32 rows x 96 params)
        #pragma unroll
        for (int jt = 0; jt < NT2; ++jt) {
            const float bias = b2[colbase + 16 * jt + m];
            #pragma unroll
            for (int mt = 0; mt < MTILES; ++mt) {
                #pragma unroll
                for (int v = 0; v < 8; ++v) {
                    const float pv = (acc[mt][jt][v] + bias) * etav;
                    pbuf[(mt * 16 + 8 * half + v) * PCHUNK + 16 * jt + m] = pv;
                }
            }
        }

        // ---- spline: lane handles (row = lane, dh = dp); full wave busy ----
        {
            const float* P  = &pbuf[lane * PCHUNK];
            const float* uw = P;
            const float* uh = P + 32;
            const float* ud = P + 64;
            const float  th = theta[(size_t)(rowbase + lane) * DH + dp];

            // pass 1: softmax maxes (stability, matches jax.nn.softmax)
            float mw = -1e30f, mh = -1e30f;
            for (int i = 0; i < KBINS; ++i) {
                mw = fmaxf(mw, uw[i]);
                mh = fmaxf(mh, uh[i]);
            }
            // pass 2: exp sums
            float sw = 0.f, sh = 0.f;
            for (int i = 0; i < KBINS; ++i) {
                sw += __expf(uw[i] - mw);
                sh += __expf(uh[i] - mh);
            }
            const float wscale = (1.0f - MIN_W * KBINS) / sw;
            const float hscale = (1.0f - MIN_H * KBINS) / sh;

            // pass 3: cumulative widths/heights + bin select (searchsorted)
            float cw = 0.f, ch = 0.f;
            float Sw = 0.f, Sh = 0.f;
            int   idx = 0;
            float sel_cw = 0.f, sel_ch = 0.f, sel_w = TWO_PI_F, sel_h = TWO_PI_F;
            for (int i = 0; i < KBINS; ++i) {
                const float wfrac = MIN_W + wscale * __expf(uw[i] - mw);
                const float hfrac = MIN_H + hscale * __expf(uh[i] - mh);
                Sw += wfrac;
                Sh += hfrac;
                const float cw_next = (i == KBINS - 1) ? TWO_PI_F : TWO_PI_F * Sw;
                const float ch_next = (i == KBINS - 1) ? TWO_PI_F : TWO_PI_F * Sh;
                if (th >= cw) {             // last true i wins
                    idx    = i;
                    sel_cw = cw;  sel_w = cw_next - cw;
                    sel_ch = ch;  sel_h = ch_next - ch;
                }
                cw = cw_next;
                ch = ch_next;
            }

            // circular derivatives: d[K] == d[0]
            const float d0 = MIN_D + softplus_f(ud[idx] + DERIV_SHIFT);
            const float d1 = MIN_D + softplus_f(ud[(idx + 1) & (KBINS - 1)] + DERIV_SHIFT);

            const float delta = sel_h / sel_w;
            const float t     = (th - sel_cw) / sel_w;
            const float t1mt  = t * (1.0f - t);
            const float numer = sel_h * (delta * t * t + d0 * t1mt);
            const float denom = delta + (d0 + d1 - 2.0f * delta) * t1mt;
            const float outv  = sel_ch + numer / denom;

            const float omt  = 1.0f - t;
            const float dnum = delta * delta * (d1 * t * t + 2.0f * delta * t1mt + d0 * omt * omt);
            const float lad  = __logf(dnum) - 2.0f * __logf(denom);

            const size_t o = (size_t)(rowbase + lane) * DH + dp;
            out[o] = outv;
            out[(size_t)Btotal * DH + o] = lad;
        }
        // pbuf is private per wave and DS ops are in-order: next dp may reuse it.
    }
}

extern "C" void kernel_launch(void* const* d_in, const int* in_sizes, int n_in,
                              void* d_out, int out_size, void* d_ws, size_t ws_size,
                              hipStream_t stream) {
    (void)n_in; (void)out_size; (void)d_ws; (void)ws_size;
    const float* theta = (const float*)d_in[0];
    const float* x     = (const float*)d_in[1];
    const float* W1    = (const float*)d_in[2];
    const float* b1    = (const float*)d_in[3];
    const float* W2    = (const float*)d_in[4];
    const float* b2    = (const float*)d_in[5];
    const float* eta   = (const float*)d_in[6];
    float* out = (float*)d_out;

    const int B = in_sizes[0] / DH;                 // 131072
    const int blocks = B / ROWS_PER_BLOCK;          // 2048
    Circular_Spline_Flow_fused_kernel<<<blocks, WAVES * 32, LDS_BYTES, stream>>>(
        theta, x, W1, b1, W2, b2, eta, out, B);
}